// CTCLossLayer_13494787244174
// MI455X (gfx1250) — compile-verified
//
#include <hip/hip_runtime.h>

#define B_   32
#define T_   2048
#define V_   128
#define L_   256
#define S_   513
#define NEGF (-1e30f)

// Branchless penalty: blank(0)/space(3) -> 1.0, vowels {11,15,19,25,31} -> 5.0.
__device__ __forceinline__ float pen_of(int v) {
    unsigned b1 = ((unsigned)v < 32u) ? ((0x00000009u >> v) & 1u) : 0u;
    unsigned b5 = ((unsigned)v < 32u) ? ((0x82088800u >> v) & 1u) : 0u;
    return (float)b1 + 5.0f * (float)b5;
}

// Low 32 bits of a flat pointer to a __shared__ object == LDS byte offset.
__device__ __forceinline__ unsigned lds_off(const void* p) {
    return (unsigned)(unsigned long long)p;
}

// CDNA5 async DMA: global -> LDS, GVS mode (SGPR base + VGPR 32-bit offset).
// Tracked by ASYNCcnt; vdst VGPR carries the LDS byte address.
__device__ __forceinline__ void async_load_b32(const float* gbase,
                                               unsigned voff_bytes,
                                               unsigned lds_addr) {
    asm volatile("global_load_async_to_lds_b32 %0, %1, %2"
                 :: "v"(lds_addr), "v"(voff_bytes), "s"(gbase)
                 : "memory");
}

__device__ __forceinline__ void wait_async0() {
    asm volatile("s_wait_asynccnt 0x0" ::: "memory");
}
// Allow 2 newest DMAs in flight; everything older has landed in LDS.
__device__ __forceinline__ void wait_async_le2() {
    asm volatile("s_wait_asynccnt 0x2" ::: "memory");
}

__device__ __forceinline__ float logadd3(float a, float a1, float a2) {
    float m = fmaxf(a, fmaxf(a1, a2));
    if (m <= -1e29f) return NEGF;
    return m + __logf(__expf(a - m) + __expf(a1 - m) + __expf(a2 - m));
}

// ---------------------------------------------------------------------------
// Kernel 1: per-(b,t) log-sum-exp of penalty-adjusted logits.
// One wave32 per row; each lane owns 4 consecutive vocab entries (b128 load).
// Fully data-parallel: 65536 rows spread across the whole chip.
// ---------------------------------------------------------------------------
__global__ __launch_bounds__(256)
void lse_kernel(const float* __restrict__ y_pred, float* __restrict__ lse_out) {
    const int lane = threadIdx.x & 31;
    const int wave = threadIdx.x >> 5;
    const int row  = blockIdx.x * 8 + wave;          // row in [0, B*T)

    const float4* rp = (const float4*)(y_pred + (size_t)row * V_);
    float4 x = rp[lane];
    const int v0 = lane * 4;
    float a0 = x.x - pen_of(v0 + 0);
    float a1 = x.y - pen_of(v0 + 1);
    float a2 = x.z - pen_of(v0 + 2);
    float a3 = x.w - pen_of(v0 + 3);

    float m = fmaxf(fmaxf(a0, a1), fmaxf(a2, a3));
#pragma unroll
    for (int d = 16; d >= 1; d >>= 1) m = fmaxf(m, __shfl_xor(m, d, 32));
    float s = __expf(a0 - m) + __expf(a1 - m) + __expf(a2 - m) + __expf(a3 - m);
#pragma unroll
    for (int d = 16; d >= 1; d >>= 1) s += __shfl_xor(s, d, 32);
    if (lane == 0) lse_out[row] = m + __logf(s);
}

// ---------------------------------------------------------------------------
// Kernel 2: serial CTC forward scan, one block per batch element.
// 544 threads = 17 waves; thread s owns alpha state s (s < 513).
// Logit rows stream in through a 4-deep async-DMA ring (s_wait_asynccnt 2
// keeps 3 steps of latency hiding); one barrier per step.
// ---------------------------------------------------------------------------
#define NT_ 544

__global__ __launch_bounds__(NT_)
void ctc_scan_kernel(const float* __restrict__ y_pred,
                     const int*   __restrict__ y_true,
                     const float* __restrict__ lse_in,
                     float*       __restrict__ losses) {
    __shared__ float lseBuf[T_];          // 8 KB: all LSEs for this batch
    __shared__ float buf[4][V_];          // 4-deep ring of logit rows
    __shared__ float alpha[2][S_ + 3];    // ping-pong alpha

    const int b   = blockIdx.x;
    const int tid = threadIdx.x;
    const bool active = (tid < S_);
    const float* base    = y_pred + (size_t)b * T_ * V_;
    const float* lseBase = lse_in + (size_t)b * T_;
    const int*   lab     = y_true + b * L_;

    // t-invariant per-state metadata: vocab index, penalty, skip flag.
    int idx0 = 0, skip0 = 0;
    if ((tid & 1) && active) {
        int l = (tid - 1) >> 1;
        idx0  = lab[l];
        skip0 = (l >= 1) && (idx0 != 0) && (idx0 != lab[l - 1]);
    }
    const float pen0 = pen_of(idx0);

    // ---- prologue: async-DMA all LSEs + rows 0..3 into the ring ----
#pragma unroll
    for (int k = 0; k < 4; ++k) {
        int i = tid + k * NT_;
        if (i < T_)
            async_load_b32(lseBase, (unsigned)i * 4u, lds_off(&lseBuf[i]));
    }
    if (tid < V_) {
#pragma unroll
        for (int r = 0; r < 4; ++r)
            async_load_b32(base, (unsigned)(r * V_ + tid) * 4u,
                           lds_off(&buf[r][tid]));
    }
    wait_async0();
    __syncthreads();

    // alpha at t=0
    float a_self = NEGF;
    if (active) {
        a_self = (tid < 2) ? (buf[0][idx0] - pen0) - lseBuf[0] : NEGF;
        alpha[0][tid] = a_self;
    }
    __syncthreads();

    // ---- serial scan: t = 1 .. T-1, one barrier per step ----
    for (int t = 1; t < T_; ++t) {
        const int cur = t & 1;
        const int nxt = cur ^ 1;

        // Keep the DMA ring 3 rows ahead of consumption.
        if (tid < V_ && (t + 3) < T_)
            async_load_b32(base, (unsigned)((t + 3) * V_ + tid) * 4u,
                           lds_off(&buf[(t + 3) & 3][tid]));

        if (active) {
            const float  lse = lseBuf[t];
            const float* row = buf[t & 3];
            const float* ap  = alpha[nxt];   // alpha[t-1] (for neighbors)

            float n1 = (tid >= 1) ? ap[tid - 1] : NEGF;
            float n2 = skip0 ? ap[tid - 2] : NEGF;
            float lp = (row[idx0] - pen0) - lse;
            a_self = logadd3(a_self, n1, n2) + lp;
            alpha[cur][tid] = a_self;
        }
        wait_async_le2();   // row t+1 (issued 3 steps ago) now resident in LDS
        __syncthreads();    // alpha[t] published; ring slot t&3 recyclable
    }

    // ---- readout: alpha[(T-1)&1] == alpha[1] ----
    if (tid == 0) {
        int len = 0;
        for (int l = 0; l < L_; ++l) len += (lab[l] != 0) ? 1 : 0;
        int i_last = 2 * len;
        int i_prev = (i_last - 1 > 0) ? (i_last - 1) : 0;
        float a  = alpha[1][i_last];
        float b2 = alpha[1][i_prev];
        float m  = fmaxf(a, b2);
        float ll = (m <= -1e29f) ? NEGF
                                 : m + __logf(__expf(a - m) + __expf(b2 - m));
        losses[b] = -ll;
    }
}

__global__ void ctc_reduce_kernel(const float* __restrict__ losses,
                                  float* __restrict__ out) {
    int lane = threadIdx.x;
    float v = (lane < B_) ? losses[lane] : 0.0f;
#pragma unroll
    for (int d = 16; d >= 1; d >>= 1) v += __shfl_xor(v, d, 32);
    if (lane == 0) out[0] = v / (float)B_ + 1e-7f;
}

extern "C" void kernel_launch(void* const* d_in, const int* in_sizes, int n_in,
                              void* d_out, int out_size, void* d_ws, size_t ws_size,
                              hipStream_t stream) {
    const float* y_pred = (const float*)d_in[0];   // [B,T,V] f32
    const int*   y_true = (const int*)d_in[1];     // [B,L]   i32

    float* lse    = (float*)d_ws;                  // B*T floats
    float* losses = lse + (size_t)B_ * T_;         // B floats
    float* out    = (float*)d_out;                 // scalar f32

    lse_kernel<<<(B_ * T_) / 8, 256, 0, stream>>>(y_pred, lse);
    ctc_scan_kernel<<<B_, NT_, 0, stream>>>(y_pred, y_true, lse, losses);
    ctc_reduce_kernel<<<1, 32, 0, stream>>>(losses, out);
}